// IPM_6141803233557
// MI455X (gfx1250) — compile-verified
//
#include <hip/hip_runtime.h>
#include <hip/hip_bf16.h>

typedef __attribute__((ext_vector_type(2))) float v2f;
typedef __attribute__((ext_vector_type(8))) float v8f;

#define NLVL 4
#define BH   100
#define BW   200
#define QTOT (NLVL*BH*BW)   /* 80000 */
#define NB   8
#define NCAM 7
#define CH   512
#define CW   896
#define EPSF 1e-9f

// One wave handles a tile of 16 BEV points. Per WMMA pass p:
//   A (16x4)  = rows 0..2  : ego2cam[b][2p]   rows 0..2
//               rows 8..10 : ego2cam[b][2p+1] rows 0..2   (others zero)
//   B (4x16)  = 16 homogeneous points (x, y, h, 1), built once
//   D (16x16) : lane l<16  -> d[0..2] = (u,v,z) of point l    for cam 2p
//               lane l>=16 -> d[0..2] = (u,v,z) of point l-16 for cam 2p+1
// so every lane owns one (point, camera) projection and samples independently.
__global__ __launch_bounds__(256) void ipm_wmma_kernel(
    const float* __restrict__ cam_feat,   // (8,7,512,896)
    const float* __restrict__ ego2cam,    // (8,7,4,4)
    float* __restrict__ out)              // (8,4,4,100,200) ++ (8,1,4,100,200)
{
    const int lane = threadIdx.x & 31;
    const int wave = threadIdx.x >> 5;
    const int b    = blockIdx.y;
    const int tile = blockIdx.x * 8 + wave;   // 0..4999
    const int q0   = tile * 16;

    const int half = lane >> 4;               // 0: K/rows 0..1 side, 1: K/rows 2..3 side
    const int l16  = lane & 15;
    const int k0   = half * 2;

    // ---- this lane's point (column of B), analytic BEV plane coords ----
    const int q   = q0 + l16;
    const int lvl = q / (BH * BW);
    const int rem = q % (BH * BW);
    const int iy  = rem / BW;
    const int ix  = rem % BW;
    const float px = -30.0f + 60.0f * (float)ix * (1.0f / 199.0f);
    const float py = -15.0f + 30.0f * (float)iy * (1.0f / 99.0f);
    const float ph = (lvl == 0) ? -1.1f : (lvl == 1) ? 0.0f : (lvl == 2) ? 0.5f : 1.1f;

    v2f Bm;                     // B[k0][n], B[k0+1][n] with n = l16
    Bm.x = half ? ph   : px;
    Bm.y = half ? 1.0f : py;

    const float* Mb = ego2cam + b * NCAM * 16;
    const float* Ib = cam_feat + (size_t)b * NCAM * (size_t)(CH * CW);

    // ---- hoist all 4 passes' A fragments: loads issue together, one wait ----
    const int  camSel = (l16 >= 8) ? 1 : 0;                 // which cam of the pair
    const int  rowIdx = (l16 < 3) ? l16 : (l16 - 8);        // matrix row 0..2
    const bool hasA   = (l16 < 3) || (l16 >= 8 && l16 < 11);

    v2f Amv[4];
#pragma unroll
    for (int p = 0; p < 4; ++p) {
        Amv[p].x = 0.0f; Amv[p].y = 0.0f;
        const int cam = 2 * p + camSel;
        if (hasA && cam < NCAM) {
            const float* Mr = Mb + cam * 16 + rowIdx * 4;
            Amv[p].x = Mr[k0];
            Amv[p].y = Mr[k0 + 1];
        }
    }

    float acc = 0.0f;
    float cnt = 0.0f;

#pragma unroll
    for (int p = 0; p < 4; ++p) {
        v8f d = {};
        d = __builtin_amdgcn_wmma_f32_16x16x4_f32(
                /*neg_a=*/false, Amv[p], /*neg_b=*/false, Bm,
                /*c_mod=*/(short)0, d, /*reuse_a=*/false, /*reuse_b=*/false);

        const int cs = 2 * p + half;     // camera this lane samples
        const float u  = d[0];
        const float vv = d[1];
        const float z  = d[2];

        float s = 0.0f, valid = 0.0f;
        if (cs < NCAM && z > EPSF) {
            // gx in (-1,1)  <=>  tx = u/z + EPS in (0, CW); sample x = tx - 0.5
            const float rz = __builtin_amdgcn_rcpf(z);
            const float tx = u  * rz + EPSF;
            const float ty = vv * rz + EPSF;
            if (tx > 0.0f && tx < (float)CW && ty > 0.0f && ty < (float)CH) {
                valid = 1.0f;
                const float sx = tx - 0.5f;
                const float sy = ty - 0.5f;
                const float fx0 = floorf(sx), fy0 = floorf(sy);
                const int   x0 = (int)fx0,   y0 = (int)fy0;   // x0,y0 in [-1, dim-1]
                // zero-padding via weight kill; loads use clamped (always valid) indices
                float wx1 = sx - fx0,  wy1 = sy - fy0;
                float wx0 = 1.0f - wx1, wy0 = 1.0f - wy1;
                const int x1 = x0 + 1, y1 = y0 + 1;            // in [0, dim]
                if (x0 < 0)      wx0 = 0.0f;
                if (x1 > CW - 1) wx1 = 0.0f;
                if (y0 < 0)      wy0 = 0.0f;
                if (y1 > CH - 1) wy1 = 0.0f;
                const int x0c = (x0 < 0) ? 0 : x0;
                const int x1c = (x1 > CW - 1) ? CW - 1 : x1;
                const int y0c = (y0 < 0) ? 0 : y0;
                const int y1c = (y1 > CH - 1) ? CH - 1 : y1;
                const float* im = Ib + (size_t)cs * (size_t)(CH * CW);
                const int r0 = y0c * CW, r1 = y1c * CW;
                // 4 unconditional loads -> issue together, one wait
                const float t00 = im[r0 + x0c];
                const float t10 = im[r0 + x1c];
                const float t01 = im[r1 + x0c];
                const float t11 = im[r1 + x1c];
                s = t00 * wx0 * wy0 + t10 * wx1 * wy0
                  + t01 * wx0 * wy1 + t11 * wx1 * wy1;
            }
        }
        acc += s;
        cnt += valid;
    }

    // merge the two half-wave camera streams for each point
    acc += __shfl_xor(acc, 16, 32);
    cnt += __shfl_xor(cnt, 16, 32);

    if (half == 0) {
        const float denom = (cnt > 0.0f) ? cnt : 1.0f;
        float* ob = out + (size_t)b * 4 * QTOT;
        ob[0 * QTOT + q] = acc / denom;
        ob[1 * QTOT + q] = px;
        ob[2 * QTOT + q] = py;
        ob[3 * QTOT + q] = ph;
        // second return value: (num > 0), appended after the (B,4,...) tensor
        out[(size_t)NB * 4 * QTOT + (size_t)b * QTOT + q] = (cnt > 0.0f) ? 1.0f : 0.0f;
    }
}

extern "C" void kernel_launch(void* const* d_in, const int* in_sizes, int n_in,
                              void* d_out, int out_size, void* d_ws, size_t ws_size,
                              hipStream_t stream) {
    (void)in_sizes; (void)n_in; (void)out_size; (void)d_ws; (void)ws_size;
    const float* cam_feat = (const float*)d_in[0];   // (8,7,512,896) f32
    const float* ego2cam  = (const float*)d_in[1];   // (8,7,4,4)     f32
    float* out = (float*)d_out;

    // 5000 tiles/batch, 8 waves (tiles) per 256-thread block -> 625 blocks x 8 batches
    dim3 grid(QTOT / 16 / 8, NB);
    ipm_wmma_kernel<<<grid, dim3(256), 0, stream>>>(cam_feat, ego2cam, out);
}